// CropAndResize_79766132621686
// MI455X (gfx1250) — compile-verified
//
#include <hip/hip_runtime.h>

// crop_and_resize, MI455X (gfx1250, wave32).
// Fixed by the reference setup: C=256, H=W=200. N and ch*cw derived at launch;
// crop_height/crop_width are device scalars read in-kernel (uniform s_loads).
#define C_CH    256
#define IMG_H   200
#define IMG_W   200
#define CGROUPS 32                 // channel-group dimension (grid.x)
#define CPG     (C_CH / CGROUPS)   // 8 channels per thread (unrolled)

__global__ __launch_bounds__(256)
void crop_resize_gather_kernel(const float* __restrict__ image,
                               const float* __restrict__ boxes,
                               const int*   __restrict__ box_ind,
                               const int*   __restrict__ p_ch,
                               const int*   __restrict__ p_cw,
                               float*       __restrict__ out)
{
    const int n  = blockIdx.y;     // box index   (uniform -> scalar loads)
    const int cg = blockIdx.x;     // base channel (uniform)
    const int ch = *p_ch;          // uniform scalar load
    const int cw = *p_cw;
    const int pixels = ch * cw;

    // Box parameters: grid-uniform -> s_load_b128 / s_load_b32.
    const float y1 = boxes[4 * n + 0];
    const float x1 = boxes[4 * n + 1];
    const float y2 = boxes[4 * n + 2];
    const float x2 = boxes[4 * n + 3];
    const int   b  = box_ind[n];

    const float hmax = (float)(IMG_H - 1);
    const float wmax = (float)(IMG_W - 1);
    const float hscale = (ch > 1) ? (y2 - y1) * hmax / (float)(ch - 1) : 0.0f;
    const float wscale = (cw > 1) ? (x2 - x1) * wmax / (float)(cw - 1) : 0.0f;
    const float y0v = y1 * hmax;
    const float x0v = x1 * wmax;

    const size_t plane   = (size_t)IMG_H * IMG_W;        // floats per channel
    const size_t istride = (size_t)CGROUPS * plane;      // channel step (32 ch)
    const float* __restrict__ basecg = image + ((size_t)b * C_CH + cg) * plane;
    float* __restrict__ outcg = out + ((size_t)n * C_CH + cg) * (size_t)pixels;
    const size_t ostride = (size_t)CGROUPS * pixels;

    // Lanes sweep crop pixels (x fastest): adjacent lanes hit the same /
    // adjacent 128B lines on taps, and stores are fully coalesced.
    for (int p = threadIdx.x; p < pixels; p += blockDim.x) {
        const int y = p / cw;            // single uniform-divisor div sequence
        const int x = p - y * cw;

        const float in_y = fmaf((float)y, hscale, y0v);
        const float in_x = fmaf((float)x, wscale, x0v);
        const bool valid = (in_y >= 0.0f) && (in_y <= hmax) &&
                           (in_x >= 0.0f) && (in_x <= wmax);

        const float yc = fminf(fmaxf(in_y, 0.0f), hmax);
        const float xc = fminf(fmaxf(in_x, 0.0f), wmax);
        const int y_lo = (int)yc;                 // yc >= 0 -> trunc == floor
        const int x_lo = (int)xc;
        const int y_hi = min(y_lo + 1, IMG_H - 1);
        const int x_hi = min(x_lo + 1, IMG_W - 1);
        const float ly = yc - (float)y_lo;
        const float lx = xc - (float)x_lo;

        const int o_tl = y_lo * IMG_W + x_lo;
        const int o_tr = y_lo * IMG_W + x_hi;
        const int o_bl = y_hi * IMG_W + x_lo;
        const int o_br = y_hi * IMG_W + x_hi;

        float* __restrict__ optr = outcg + p;

        if (!valid) {
            // Extrapolation value for all 8 channels of this pixel.
#pragma unroll
            for (int i = 0; i < CPG; ++i) {
                __builtin_nontemporal_store(0.0f, optr);
                optr += ostride;
            }
            continue;
        }

        // 8-deep channel unroll: 32 independent global_load_b32 in flight
        // per thread; interpolation weights are channel-invariant. Output is
        // write-once -> non-temporal stores (th:NT) keep the 103 MB output
        // stream out of the 192 MB L2, preserving residency for image-line
        // reuse across the ~64 overlapping boxes per source image.
        const float* __restrict__ bp = basecg;
#pragma unroll
        for (int i = 0; i < CPG; ++i) {
            const float tl = bp[o_tl];
            const float tr = bp[o_tr];
            const float bl = bp[o_bl];
            const float br = bp[o_br];
            const float top = fmaf(tr - tl, lx, tl);
            const float bot = fmaf(br - bl, lx, bl);
            const float val = fmaf(bot - top, ly, top);
            __builtin_nontemporal_store(val, optr);
            bp   += istride;
            optr += ostride;
        }
    }
}

extern "C" void kernel_launch(void* const* d_in, const int* in_sizes, int n_in,
                              void* d_out, int out_size, void* d_ws, size_t ws_size,
                              hipStream_t stream) {
    const float* image   = (const float*)d_in[0];
    const float* boxes   = (const float*)d_in[1];
    const int*   box_ind = (const int*)d_in[2];
    const int*   p_ch    = (const int*)d_in[3];   // device scalar (crop_height)
    const int*   p_cw    = (const int*)d_in[4];   // device scalar (crop_width)
    float*       out     = (float*)d_out;

    const int N = in_sizes[2];                    // 512 boxes

    dim3 block(256);
    dim3 grid(CGROUPS, N);                        // (channel groups, boxes)
    crop_resize_gather_kernel<<<grid, block, 0, stream>>>(
        image, boxes, box_ind, p_ch, p_cw, out);
}